// CRF_12979391169105
// MI455X (gfx1250) — compile-verified
//
#include <hip/hip_runtime.h>
#include <hip/hip_bf16.h>

// ---------------- problem constants ----------------
#define L_LEN      32768
#define D_DIM      1024
#define C_TAGS     50
#define CP         64        // padded tag stride
#define START_TAG  48
#define STOP_TAG   49
#define CHUNK      32
#define NB         (L_LEN / CHUNK)   // 1024
#define NEG_BIG    (-1e30f)

typedef __attribute__((ext_vector_type(2))) float v2f;
typedef __attribute__((ext_vector_type(8))) float v8f;
typedef __attribute__((ext_vector_type(4))) unsigned int v4u;
typedef __attribute__((ext_vector_type(8))) unsigned int v8u;

// ============================================================
// Kernel 1: emit = features @ W^T + b  (fp32 WMMA 16x16x4)
// grid = L/16 blocks, 128 threads (4 waves); wave w owns N-tile w.
// Branchless B loads: invalid tag rows clamp to row 0; the garbage
// only reaches emit columns >= 50 which are never read downstream.
// ============================================================
__global__ void crf_emit_gemm(const float* __restrict__ F,
                              const float* __restrict__ W,
                              const float* __restrict__ bias,
                              float* __restrict__ emitW) {
    const int wave = threadIdx.x >> 5;
    const int lane = threadIdx.x & 31;
    const int half = lane >> 4;          // 0: lanes 0-15 (K=0,1), 1: lanes 16-31 (K=2,3)
    const int lr   = lane & 15;
    const int m0   = blockIdx.x * 16;
    const int n0   = wave * 16;

    const int arow  = m0 + lr;                     // feature row (A, M = lr)
    const int brow  = n0 + lr;                     // W row (B, N = lr)
    const int browc = (brow < C_TAGS) ? brow : 0;  // clamp -> branchless load

    const float* Fp = F + (size_t)arow  * D_DIM + 2 * half;
    const float* Wp = W + (size_t)browc * D_DIM + 2 * half;

    v8f acc = {};
    for (int k = 0; k < D_DIM; k += 4) {
        v2f a  = *(const v2f*)(Fp + k);            // A: M=lr, K = k+2*half..+1
        v2f bb = *(const v2f*)(Wp + k);            // B: N=lr, K = k+2*half..+1
        acc = __builtin_amdgcn_wmma_f32_16x16x4_f32(
            /*neg_a=*/false, a, /*neg_b=*/false, bb,
            /*c_mod=*/(short)0, acc, /*reuse_a=*/false, /*reuse_b=*/false);
    }

    const int col = n0 + lr;                        // D: N per lane
    const float bv = (col < C_TAGS) ? bias[col] : 0.f;
    #pragma unroll
    for (int r = 0; r < 8; ++r) {
        const int row = m0 + r + 8 * half;          // D: VGPR r -> M = r / 8+r
        emitW[(size_t)row * CP + col] = acc[r] + bv;
    }
}

// ============================================================
// Kernel 2: per-block max-plus matrix products P_b (50x50), CHUNK steps.
// T matrix (2500 f32, 10 KB) is DMA'd into LDS via the Tensor Data Mover.
// grid = NB, 128 threads
// ============================================================
__global__ void crf_block_products(const float* __restrict__ emitW,
                                   const float* __restrict__ T,
                                   float* __restrict__ blockP) {
    __shared__ float Ts[C_TAGS * C_TAGS];   // contiguous: TDM writes tile densely
    __shared__ float P0[C_TAGS * 52];
    __shared__ float P1[C_TAGS * 52];
    __shared__ float erow[CP];

    const int tid = threadIdx.x;         // 128
    const int b   = blockIdx.x;
    const int t0  = b * CHUNK;

    // ---- TDM: tensor_load_to_lds of the full 50x50 f32 T matrix ----
    if (tid < 32) {                      // one wave issues (TDM ignores EXEC)
        const unsigned lds_off = (unsigned)(unsigned long long)(const void*)&Ts[0];
        const unsigned long long ga = (unsigned long long)T;
        v4u g0;
        g0.x = 1u;                                           // count=1, user mode
        g0.y = lds_off;                                      // lds_addr
        g0.z = (unsigned)(ga & 0xffffffffull);               // global_addr[31:0]
        g0.w = (unsigned)((ga >> 32) & 0x01ffffffull)        // global_addr[56:32]
             | (2u << 30);                                   // type=2 ("image")
        v8u g1;
        g1.s0 = (2u << 16);                                  // data_size=4B
        g1.s1 = (2500u & 0xffffu) << 16;                     // tensor_dim0 lo16
        g1.s2 = (2500u >> 16) | (1u << 16);                  // dim0 hi16 | tensor_dim1=1
        g1.s3 = (2500u << 16);                               // tile_dim0 = 2500
        g1.s4 = 1u;                                          // tile_dim1=1, tile_dim2=0
        g1.s5 = 2500u;                                       // tensor_dim0_stride lo32
        g1.s6 = 0u;
        g1.s7 = 0u;
        asm volatile("tensor_load_to_lds %0, %1"
                     :: "s"(g0), "s"(g1) : "memory");
    }
    if (tid < C_TAGS) erow[tid] = emitW[(size_t)t0 * CP + tid];
    __builtin_amdgcn_s_wait_tensorcnt(0);
    __syncthreads();

    // P = M_{t0}:  M_t[i][j] = T[i][j] + emit_t[i]
    for (int idx = tid; idx < C_TAGS * C_TAGS; idx += 128) {
        int i = idx / C_TAGS, j = idx % C_TAGS;
        P0[i * 52 + j] = Ts[i * C_TAGS + j] + erow[i];
    }
    __syncthreads();

    float* Pc = P0;
    float* Pn = P1;
    for (int t = t0 + 1; t < t0 + CHUNK; ++t) {
        if (tid < C_TAGS) erow[tid] = emitW[(size_t)t * CP + tid];
        __syncthreads();
        for (int idx = tid; idx < C_TAGS * C_TAGS; idx += 128) {
            int i = idx / C_TAGS, j = idx % C_TAGS;
            const float* trow = &Ts[i * C_TAGS];
            float m = NEG_BIG;
            for (int k = 0; k < C_TAGS; ++k)
                m = fmaxf(m, trow[k] + Pc[k * 52 + j]);
            Pn[i * 52 + j] = m + erow[i];
        }
        __syncthreads();
        float* tmp = Pc; Pc = Pn; Pn = tmp;
    }
    for (int idx = tid; idx < C_TAGS * C_TAGS; idx += 128) {
        int i = idx / C_TAGS, j = idx % C_TAGS;
        blockP[(size_t)b * (CP * CP) + i * CP + j] = Pc[i * 52 + j];
    }
}

// ============================================================
// Kernel 3: sequential scan of block products -> fv checkpoints
// grid = 1, 64 threads
// ============================================================
__global__ void crf_scan_blocks(const float* __restrict__ blockP,
                                float* __restrict__ fvcp) {
    __shared__ float fv[CP];
    __shared__ float fvn[CP];
    const int tid = threadIdx.x;         // 64
    fv[tid] = (tid == START_TAG) ? 0.f : NEG_BIG;
    fvcp[tid] = fv[tid];
    __syncthreads();

    for (int b = 0; b < NB; ++b) {
        if (b + 1 < NB && tid < C_TAGS)
            __builtin_prefetch(blockP + (size_t)(b + 1) * (CP * CP) + tid * CP, 0, 0);
        if (tid < C_TAGS) {
            const float* row = blockP + (size_t)b * (CP * CP) + tid * CP;
            float m = NEG_BIG;
            for (int j = 0; j < C_TAGS; ++j)
                m = fmaxf(m, row[j] + fv[j]);
            fvn[tid] = m;
        }
        __syncthreads();
        float v = (tid < C_TAGS) ? fvn[tid] : NEG_BIG;
        fv[tid] = v;
        fvcp[(size_t)(b + 1) * CP + tid] = v;
        __syncthreads();
    }
}

// ============================================================
// Kernel 4: per-block replay from checkpoint -> backpointers
// grid = NB, 64 threads
// ============================================================
__global__ void crf_replay(const float* __restrict__ emitW,
                           const float* __restrict__ T,
                           const float* __restrict__ fvcp,
                           unsigned char* __restrict__ bps) {
    __shared__ float Ts[C_TAGS * 52];
    __shared__ float fv[52], fvn[52], erow[52];
    const int tid = threadIdx.x;         // 64
    const int b   = blockIdx.x;
    const int t0  = b * CHUNK;

    for (int idx = tid; idx < C_TAGS * C_TAGS; idx += 64)
        Ts[(idx / C_TAGS) * 52 + (idx % C_TAGS)] = T[idx];
    if (tid < 52) fv[tid] = (tid < C_TAGS) ? fvcp[(size_t)b * CP + tid] : NEG_BIG;
    __syncthreads();

    for (int t = t0; t < t0 + CHUNK; ++t) {
        if (tid < C_TAGS) erow[tid] = emitW[(size_t)t * CP + tid];
        __syncthreads();
        if (tid < C_TAGS) {
            const float* trow = &Ts[tid * 52];
            float best = NEG_BIG;
            int bj = 0;
            for (int j = 0; j < C_TAGS; ++j) {
                float s = fv[j] + trow[j];
                if (s > best) { best = s; bj = j; }   // first-max, matches jnp.argmax
            }
            fvn[tid] = best + erow[tid];
            bps[(size_t)t * CP + tid] = (unsigned char)bj;
        }
        __syncthreads();
        if (tid < C_TAGS) fv[tid] = fvn[tid];
        __syncthreads();
    }
}

// ============================================================
// Kernel 5: terminal score + best tag
// grid = 1, 64 threads
// ============================================================
__global__ void crf_terminal(const float* __restrict__ fvcp,
                             const float* __restrict__ T,
                             float* __restrict__ out,
                             int* __restrict__ bestTag) {
    __shared__ float term[CP];
    __shared__ int   idx[CP];
    const int tid = threadIdx.x;         // 64
    float v = NEG_BIG;
    if (tid < C_TAGS)
        v = fvcp[(size_t)NB * CP + tid] + T[STOP_TAG * C_TAGS + tid];
    term[tid] = v;
    idx[tid]  = tid;
    __syncthreads();
    for (int s = 32; s > 0; s >>= 1) {
        if (tid < s) {
            if (term[tid + s] > term[tid]) {   // ties keep lower index
                term[tid] = term[tid + s];
                idx[tid]  = idx[tid + s];
            }
        }
        __syncthreads();
    }
    if (tid == 0) { out[0] = term[0]; *bestTag = idx[0]; }
}

// ============================================================
// Kernel 6: per-block composed backtrack maps F_b : tag@top -> tag below block
// grid = NB, 64 threads
// ============================================================
__global__ void crf_block_maps(const unsigned char* __restrict__ bps,
                               unsigned char* __restrict__ Fmap) {
    __shared__ unsigned char lb[CHUNK * CP];
    const int tid = threadIdx.x;         // 64
    const int b   = blockIdx.x;
    const int t0  = b * CHUNK;
    for (int i = tid; i < CHUNK * CP; i += 64)
        lb[i] = bps[(size_t)t0 * CP + i];
    __syncthreads();
    if (tid < C_TAGS) {
        int cur = tid;
        for (int tt = CHUNK - 1; tt >= 0; --tt)
            cur = lb[tt * CP + cur];
        Fmap[(size_t)b * CP + tid] = (unsigned char)cur;
    }
}

// ============================================================
// Kernel 7: sequential compose of the NB block maps -> boundary tags
// grid = 1, 32 threads (lane 0 works; 1024 dependent L2 hits)
// ============================================================
__global__ void crf_compose(const unsigned char* __restrict__ Fmap,
                            const int* __restrict__ bestTag,
                            int* __restrict__ topTag) {
    if (threadIdx.x == 0) {
        int cur = *bestTag;
        topTag[NB - 1] = cur;
        for (int b = NB - 1; b >= 1; --b) {
            cur = Fmap[(size_t)b * CP + cur];
            topTag[b - 1] = cur;
        }
    }
}

// ============================================================
// Kernel 8: parallel path fill (one thread per block) + transitions copy
// ============================================================
__global__ void crf_fill_path(const unsigned char* __restrict__ bps,
                              const int* __restrict__ topTag,
                              float* __restrict__ out) {
    const int b = blockIdx.x * blockDim.x + threadIdx.x;
    if (b < NB) {
        const int t0 = b * CHUNK;
        int cur = topTag[b];
        out[1 + t0 + CHUNK - 1] = (float)cur;
        for (int t = t0 + CHUNK - 1; t > t0; --t) {
            cur = bps[(size_t)t * CP + cur];
            out[1 + t - 1] = (float)cur;
        }
    }
}

__global__ void crf_copy_T(const float* __restrict__ T, float* __restrict__ out) {
    const int i = blockIdx.x * blockDim.x + threadIdx.x;
    if (i < C_TAGS * C_TAGS) out[1 + L_LEN + i] = T[i];
}

// ============================================================
// Host launcher
// ============================================================
extern "C" void kernel_launch(void* const* d_in, const int* in_sizes, int n_in,
                              void* d_out, int out_size, void* d_ws, size_t ws_size,
                              hipStream_t stream) {
    (void)in_sizes; (void)n_in; (void)out_size; (void)ws_size;

    const float* F    = (const float*)d_in[0];   // [L, D]
    const float* W    = (const float*)d_in[1];   // [C, D]
    const float* bias = (const float*)d_in[2];   // [C]
    const float* T    = (const float*)d_in[3];   // [C, C]
    float* out        = (float*)d_out;           // [1 + L + C*C]

    // workspace layout
    char* ws = (char*)d_ws;
    float* emitW          = (float*)ws;                                  // L*CP*4  = 8 MB
    float* blockP         = (float*)(ws + ((size_t)8 << 20));            // NB*CP*CP*4 = 16 MB
    float* fvcp           = (float*)(ws + ((size_t)24 << 20));           // (NB+1)*CP*4
    unsigned char* bps    = (unsigned char*)(ws + ((size_t)24 << 20) + (512u << 10)); // L*CP = 2 MB
    unsigned char* Fmap   = bps + (size_t)L_LEN * CP;                    // NB*CP
    int* topTag           = (int*)(Fmap + (size_t)NB * CP);              // NB ints
    int* bestTag          = topTag + NB;

    // 1) emit GEMM (fp32 WMMA)
    crf_emit_gemm<<<L_LEN / 16, 128, 0, stream>>>(F, W, bias, emitW);
    // 2) block max-plus matrix products (TDM-fed T matrix)
    crf_block_products<<<NB, 128, 0, stream>>>(emitW, T, blockP);
    // 3) scan across blocks -> fv checkpoints
    crf_scan_blocks<<<1, 64, 0, stream>>>(blockP, fvcp);
    // 4) replay within blocks -> backpointers
    crf_replay<<<NB, 64, 0, stream>>>(emitW, T, fvcp, bps);
    // 5) terminal score / best tag
    crf_terminal<<<1, 64, 0, stream>>>(fvcp, T, out, bestTag);
    // 6) per-block composed backtrack maps
    crf_block_maps<<<NB, 64, 0, stream>>>(bps, Fmap);
    // 7) sequential compose of maps -> block-boundary tags
    crf_compose<<<1, 32, 0, stream>>>(Fmap, bestTag, topTag);
    // 8) parallel path fill + transitions copy
    crf_fill_path<<<NB / 32, 32, 0, stream>>>(bps, topTag, out);
    crf_copy_T<<<(C_TAGS * C_TAGS + 127) / 128, 128, 0, stream>>>(T, out);
}